// ModelMetricConsistency_44109314130012
// MI455X (gfx1250) — compile-verified
//
#include <hip/hip_runtime.h>

typedef __attribute__((ext_vector_type(2))) float v2f;
typedef __attribute__((ext_vector_type(8))) float v8f;

#define NT  8            // 16-column gt subtiles per wave (128 gt points / wave)
#define WPB 4            // waves per block
#define TPB_CH (WPB * 32)

// ---------------------------------------------------------------------------
// Init: per-(b,m) min buffer to +inf bit pattern
// ---------------------------------------------------------------------------
__global__ void init_minm_kernel(unsigned* __restrict__ minm, int total) {
  int i = blockIdx.x * blockDim.x + threadIdx.x;
  if (i < total) minm[i] = 0x7F800000u;  // +inf
}

// ---------------------------------------------------------------------------
// Chamfer via V_WMMA_F32_16X16X4_F32:
//   A (16x4): rows = pred points, K = {x, y, z, ||p||^2}
//   B (4x16): cols = gt points,  K = {-2gx, -2gy, -2gz, 1}
//   D = ||p||^2 - 2 p.g ;  dist^2 = D + ||g||^2
// Each wave: fixed 128 gt columns, loops all pred 16-row tiles.
//   - min over m for each n: finished in-wave (registers + shfl), partial sums stored
//   - min over n for each m: combined across waves with uint-bit atomicMin
// ---------------------------------------------------------------------------
__global__ __launch_bounds__(TPB_CH)
void chamfer_wmma_kernel(const float* __restrict__ pc_gt,
                         const float* __restrict__ pc_pred,
                         unsigned* __restrict__ minm,
                         float* __restrict__ gt2p_part,
                         int B, int N, int M, int mpad) {
  const int lane = threadIdx.x & 31;
  const int wave = threadIdx.x >> 5;
  const int b    = blockIdx.x;
  const int l16  = lane & 15;
  const int half = lane >> 4;
  const int widx   = blockIdx.y * WPB + wave;
  const int n_base = widx * (NT * 16);
  const int nwaves = gridDim.y * WPB;

  // Load B operands (gt columns) once; 16x16x4 f32 B layout:
  // lanes 0-15: vgpr0=K0, vgpr1=K1 ; lanes 16-31: vgpr0=K2, vgpr1=K3
  v2f   Bt[NT];
  float gn2[NT];
  float minN[NT];
#pragma unroll
  for (int t = 0; t < NT; t++) {
    int n = n_base + t * 16 + l16;
    const float* g = pc_gt + ((size_t)b * N + (size_t)n) * 3;
    float gx = g[0], gy = g[1], gz = g[2];
    gn2[t] = gx * gx + gy * gy + gz * gz;
    v2f bb;
    bb.x = half ? (-2.0f * gz) : (-2.0f * gx);
    bb.y = half ? 1.0f         : (-2.0f * gy);
    Bt[t]   = bb;
    minN[t] = 3.0e38f;
  }

  const int mtiles = (M + 15) >> 4;
  for (int mt = 0; mt < mtiles; mt++) {
    // A operand: lanes 0-15: {x,y} of row l16 ; lanes 16-31: {z,||p||^2}
    int m = mt * 16 + l16;
    float x = 0.0f, y = 0.0f, z = 0.0f, pn2 = 3.0e37f;  // pad rows -> huge dist
    if (m < M) {
      const float* p = pc_pred + ((size_t)b * M + (size_t)m) * 3;
      x = p[0]; y = p[1]; z = p[2];
      pn2 = x * x + y * y + z * z;
    }
    v2f A;
    A.x = half ? z   : x;
    A.y = half ? pn2 : y;

    float rowmin[8];
#pragma unroll
    for (int v = 0; v < 8; v++) rowmin[v] = 3.0e38f;

#pragma unroll
    for (int t = 0; t < NT; t++) {
      v8f acc = {};
      acc = __builtin_amdgcn_wmma_f32_16x16x4_f32(
          /*neg_a=*/false, A, /*neg_b=*/false, Bt[t],
          /*c_mod=*/(short)0, acc, /*reuse_a=*/false, /*reuse_b=*/false);
#pragma unroll
      for (int v = 0; v < 8; v++) {
        float d2 = acc[v] + gn2[t];          // squared distance (maybe ~-eps)
        minN[t]   = fminf(minN[t], d2);      // min over m for this lane's n
        rowmin[v] = fminf(rowmin[v], d2);    // min over n for row m
      }
    }
    // Reduce rowmin across the 16 lanes of each half (D: N striped over lanes)
#pragma unroll
    for (int v = 0; v < 8; v++) {
      float r = rowmin[v];
      r = fminf(r, __shfl_xor(r, 1, 32));
      r = fminf(r, __shfl_xor(r, 2, 32));
      r = fminf(r, __shfl_xor(r, 4, 32));
      r = fminf(r, __shfl_xor(r, 8, 32));
      rowmin[v] = fmaxf(r, 0.0f);            // clamp -> nonneg, uint order == float order
    }
    // D row mapping: half 0 -> m = mt*16 + v ; half 1 -> m = mt*16 + 8 + v
#pragma unroll
    for (int v = 0; v < 8; v++) {
      int mo = mt * 16 + half * 8 + v;
      if (l16 == v && mo < M) {
        atomicMin(&minm[(size_t)b * mpad + mo], __float_as_uint(rowmin[v]));
      }
    }
  }

  // gt->pred: combine halves (lane L and L+16 share the same n), sum this
  // wave's 128 column-mins, store deterministic per-wave partial.
  float s = 0.0f;
#pragma unroll
  for (int t = 0; t < NT; t++) {
    float mn = fminf(minN[t], __shfl_xor(minN[t], 16, 32));
    mn = fmaxf(mn, 0.0f);
    if (half == 0) s += mn;
  }
  s += __shfl_xor(s, 16, 32);
  s += __shfl_xor(s, 8, 32);
  s += __shfl_xor(s, 4, 32);
  s += __shfl_xor(s, 2, 32);
  s += __shfl_xor(s, 1, 32);
  if (lane == 0) gt2p_part[(size_t)b * nwaves + widx] = s;
}

// ---------------------------------------------------------------------------
// Finisher: chamfer sums + all FFF losses -> scalar. Single 256-thread block,
// fixed LDS reduction trees (deterministic).
// ---------------------------------------------------------------------------
__device__ __forceinline__ float block_reduce_sum(float v, float* red) {
  const int t = threadIdx.x;
  __syncthreads();
  red[t] = v;
  __syncthreads();
#pragma unroll
  for (int s = 128; s > 0; s >>= 1) {
    if (t < s) red[t] += red[t + s];
    __syncthreads();
  }
  return red[0];
}

__global__ __launch_bounds__(256)
void loss_final_kernel(const float* __restrict__ fff,
                       const float* __restrict__ A_gt,
                       const float* __restrict__ minm_f,
                       const float* __restrict__ gt2p_part,
                       float* __restrict__ out,
                       int B, int M, int N, int mpad, int nwaves) {
  __shared__ float red[256];
  __shared__ float sh[24];
  const int tid = threadIdx.x;
  const int BM  = B * M;

  // Pass 1: E/G means, p2gt min-sum, gt2p partial-sum
  float sE = 0, sG = 0, sP = 0, sQ = 0;
  for (int i = tid; i < BM; i += 256) {
    const float* f = fff + (size_t)i * 3;
    sE += f[0];
    sG += f[2];
    int bb = i / M;
    int mm = i - bb * M;
    sP += minm_f[(size_t)bb * mpad + mm];
  }
  const int nPart = B * nwaves;
  for (int i = tid; i < nPart; i += 256) sQ += gt2p_part[i];

  float tE = block_reduce_sum(sE, red);
  float tG = block_reduce_sum(sG, red);
  float tP = block_reduce_sum(sP, red);
  float tQ = block_reduce_sum(sQ, red);
  if (tid == 0) { sh[0] = tE / (float)BM; sh[1] = tG / (float)BM; sh[2] = tP; sh[3] = tQ; }
  __syncthreads();
  const float Emean = sh[0], Gmean = sh[1];

  // Pass 2: scaled-isometry terms, metric consistency, per-batch area sums
  float sStr = 0, sE2 = 0, sG2 = 0, sF2 = 0, sMC = 0;
  for (int bb = 0; bb < B; bb++) {
    float sA = 0;
    for (int mm = tid; mm < M; mm += 256) {
      const float* f = fff + ((size_t)bb * M + mm) * 3;
      float E = f[0], F = f[1], G = f[2];
      float A2  = fmaxf(E * G - F * F, 0.0f);
      float inv = 1.0f / (A2 + 1e-20f);
      float dEG = E - G;          sStr += dEG * dEG * inv;
      float dE  = E - Emean;      sE2  += dE * dE * inv;
      float dG  = G - Gmean;      sG2  += dG * dG * inv;
      sF2 += F * F * inv;
      sA  += sqrtf(A2);
      if ((bb & 1) == 0) {        // consecutive frame pair (bb, bb+1)
        const float* f1 = fff + ((size_t)(bb + 1) * M + mm) * 3;
        float mE = E - f1[0], mF = F - f1[1], mG = G - f1[2];
        sMC += mE * mE + 2.0f * mF * mF + mG * mG;
      }
    }
    float tA = block_reduce_sum(sA, red);
    if (tid == 0) sh[4 + bb] = tA;
  }
  float tStr = block_reduce_sum(sStr, red);
  float tE2  = block_reduce_sum(sE2, red);
  float tG2  = block_reduce_sum(sG2, red);
  float tF2  = block_reduce_sum(sF2, red);
  float tMC  = block_reduce_sum(sMC, red);

  if (tid == 0) {
    const float BMf  = (float)BM;
    const float L_chd = sh[2] / BMf + sh[3] / ((float)B * (float)N);
    const float L_mc  = tMC / (0.5f * BMf);          // mean over (B/2, P, spp)
    const float sppf  = (float)(M / 10);             // P = 10
    float L_olap = 0.0f;
    for (int bb = 0; bb < B; bb++) {
      float Atot = sh[4 + bb] / sppf;                // mean over spp, sum over P
      float d = fmaxf(Atot - A_gt[bb], 0.0f);        // total_area_mult = 1
      L_olap += d * d;
    }
    L_olap /= (float)B;
    const float sciso = tF2 / BMf + tE2 / BMf + tG2 / BMf + tStr / BMf + L_olap;
    out[0] = L_chd + L_mc + sciso;                   // ALPHA_MC = ALPHA_DISTORT = 1
  }
}

// ---------------------------------------------------------------------------
extern "C" void kernel_launch(void* const* d_in, const int* in_sizes, int n_in,
                              void* d_out, int out_size, void* d_ws, size_t ws_size,
                              hipStream_t stream) {
  (void)n_in; (void)out_size; (void)ws_size;
  const float* pc_gt   = (const float*)d_in[0];
  const float* pc_pred = (const float*)d_in[1];
  const float* fff     = (const float*)d_in[2];
  const float* A_gt    = (const float*)d_in[3];
  float* out = (float*)d_out;

  const int B = in_sizes[3];               // 8
  const int N = in_sizes[0] / (3 * B);     // 8192
  const int M = in_sizes[1] / (3 * B);     // 2500
  const int mpad   = ((M + 15) / 16) * 16; // 2512
  const int nwaves = N / (NT * 16);        // waves per batch = 64
  const int blocksY = nwaves / WPB;        // 16

  // Workspace layout: [gt2p partials][pad][minm bits]
  float*    gt2p_part = (float*)d_ws;
  size_t    partBytes = (((size_t)B * nwaves * sizeof(float)) + 255) & ~(size_t)255;
  unsigned* minm      = (unsigned*)((char*)d_ws + partBytes);

  const int total = B * mpad;
  hipLaunchKernelGGL(init_minm_kernel, dim3((total + 255) / 256), dim3(256), 0, stream,
                     minm, total);
  hipLaunchKernelGGL(chamfer_wmma_kernel, dim3(B, blocksY), dim3(TPB_CH), 0, stream,
                     pc_gt, pc_pred, minm, gt2p_part, B, N, M, mpad);
  hipLaunchKernelGGL(loss_final_kernel, dim3(1), dim3(256), 0, stream,
                     fff, A_gt, (const float*)minm, gt2p_part, out, B, M, N, mpad, nwaves);
}